// GCN_83683142795703
// MI455X (gfx1250) — compile-verified
//
#include <hip/hip_runtime.h>
#include <hip/hip_bf16.h>

typedef __attribute__((ext_vector_type(2))) float v2f;
typedef __attribute__((ext_vector_type(8))) float v8f;

#define D_FEAT 64

// ---------------------------------------------------------------------------
// Kernel 0: zero the aggregation buffer (harness poisons ws with 0xAA)
// ---------------------------------------------------------------------------
__global__ void gcn_zero_kernel(float4* __restrict__ p, int n4) {
    int i = blockIdx.x * blockDim.x + threadIdx.x;
    if (i < n4) p[i] = make_float4(0.f, 0.f, 0.f, 0.f);
}

// ---------------------------------------------------------------------------
// Kernel 1: edge scatter-add.  16 threads per edge, float4 payload each.
// features gather and h_agg atomics are both L2-resident (25.6 MB each,
// 192 MB L2), so atomics resolve in the L2 atomic units.
// ---------------------------------------------------------------------------
__global__ void gcn_scatter_kernel(const float* __restrict__ features,
                                   const int* __restrict__ src,
                                   const int* __restrict__ dst,
                                   float* __restrict__ h_agg,
                                   int n_edges) {
    int gid = blockIdx.x * blockDim.x + threadIdx.x;
    int e = gid >> 4;                // edge index
    int c = (gid & 15) << 2;         // feature offset (0,4,...,60)
    if (e >= n_edges) return;
    int s = src[e];
    int d = dst[e];
    const float4 v = *(const float4*)(features + (size_t)s * D_FEAT + c);
    float* o = h_agg + (size_t)d * D_FEAT + c;
    atomicAdd(o + 0, v.x);
    atomicAdd(o + 1, v.y);
    atomicAdd(o + 2, v.z);
    atomicAdd(o + 3, v.w);
}

// ---------------------------------------------------------------------------
// Kernel 2: h = relu(h_agg @ W^T + b) via V_WMMA_F32_16X16X4_F32.
// One wave -> one 16x16 output tile; 8 waves/block -> 32 rows x 64 cols.
// K = 64 -> 16 WMMA steps.  100000 % 32 == 0, so no tails.
// ---------------------------------------------------------------------------
__global__ void __launch_bounds__(256)
gcn_gemm_relu_kernel(const float* __restrict__ h_agg,
                     const float* __restrict__ W,     // [64 out][64 in] row-major
                     const float* __restrict__ bias,  // [64]
                     float* __restrict__ out) {       // [N][64]
    const int lane = threadIdx.x & 31;
    const int wave = threadIdx.x >> 5;           // 0..7
    const int m_tile = blockIdx.x * 2 + (wave >> 2);
    const int n_tile = wave & 3;
    const int half = lane >> 4;                  // 0: K={k,k+1}, 1: K={k+2,k+3}
    const int l16  = lane & 15;
    const int m = m_tile * 16 + l16;             // A row held by this lane
    const int n = n_tile * 16 + l16;             // B col / D col held by this lane

    const float* arow = h_agg + (size_t)m * D_FEAT;
    const float* brow = W     + (size_t)n * D_FEAT;   // B[k][n] = W[n][k]

    v8f acc = {};
#pragma unroll
    for (int k = 0; k < D_FEAT; k += 4) {
        const int kk = k + half * 2;
        v2f a = *(const v2f*)(arow + kk);
        v2f b = *(const v2f*)(brow + kk);
        acc = __builtin_amdgcn_wmma_f32_16x16x4_f32(
            /*neg_a=*/false, a, /*neg_b=*/false, b,
            /*c_mod=*/(short)0, acc, /*reuse_a=*/false, /*reuse_b=*/false);
    }

    const float bv = bias[n];
#pragma unroll
    for (int r = 0; r < 8; ++r) {
        const int mm = m_tile * 16 + r + half * 8;   // D row for VGPR r
        float v = acc[r] + bv;
        out[(size_t)mm * D_FEAT + n] = v > 0.f ? v : 0.f;
    }
}

// ---------------------------------------------------------------------------
extern "C" void kernel_launch(void* const* d_in, const int* in_sizes, int n_in,
                              void* d_out, int out_size, void* d_ws, size_t ws_size,
                              hipStream_t stream) {
    const float* features = (const float*)d_in[0];   // [N, 64]
    const int*   src      = (const int*)d_in[1];     // [E]
    const int*   dst      = (const int*)d_in[2];     // [E]
    const float* W        = (const float*)d_in[3];   // [64, 64]
    const float* bias     = (const float*)d_in[4];   // [64]
    float*       out      = (float*)d_out;           // [N, 64]
    float*       h_agg    = (float*)d_ws;            // [N, 64] scratch

    const int n_nodes = in_sizes[0] / D_FEAT;        // 100000
    const int n_edges = in_sizes[1];                 // 1600000
    const int n_elems = n_nodes * D_FEAT;            // 6.4M

    // 1) zero h_agg
    {
        int n4 = n_elems / 4;
        int blocks = (n4 + 255) / 256;
        gcn_zero_kernel<<<blocks, 256, 0, stream>>>((float4*)h_agg, n4);
    }
    // 2) scatter-add over edges (16 threads / edge)
    {
        long long work = (long long)n_edges * 16;
        int blocks = (int)((work + 255) / 256);
        gcn_scatter_kernel<<<blocks, 256, 0, stream>>>(features, src, dst, h_agg, n_edges);
    }
    // 3) WMMA GEMM + bias + ReLU
    {
        int blocks = n_nodes / 32;                   // 3125, exact
        gcn_gemm_relu_kernel<<<blocks, 256, 0, stream>>>(h_agg, W, bias, out);
    }
}